// EvenOddWilsonDslash_78469052498770
// MI455X (gfx1250) — compile-verified
//
#include <hip/hip_runtime.h>
#include <cstdint>

// Even-odd Wilson Dslash, 32^4 lattice, 4 spins x 3 colors, complex64.
// hop(0->1)+hop(1->0) with mass=0 == full Wilson hopping term at every site.
// Memory-bound: ~480 MB min HBM traffic -> ~21us @ 23.3 TB/s. Spin-projection
// trick halves SU(3) work to stay under the fp32 VALU roofline.
// CDNA5 paths: ALL gauge-link traffic is staged via 4 TDM tensor_load_to_lds
// DMAs per block (1 contiguous main tile + 3 strided backward-link tiles),
// spinor loads are b128 streams warmed by global_prefetch during the DMA,
// wave32-sized blocks, s_wait_tensorcnt + barrier for the LDS handoff.

#define LSZ 32
#define VOL (LSZ * LSZ * LSZ * LSZ)
#define TPB 128   // 4 wave32s; 4 full x-rows per block; uniform t,z per block

typedef float2 cplx;

__device__ __forceinline__ cplx c_add(cplx a, cplx b) { return make_float2(a.x + b.x, a.y + b.y); }
__device__ __forceinline__ cplx c_sub(cplx a, cplx b) { return make_float2(a.x - b.x, a.y - b.y); }
__device__ __forceinline__ cplx c_i (cplx a) { return make_float2(-a.y,  a.x); }  //  i*a
__device__ __forceinline__ cplx c_mi(cplx a) { return make_float2( a.y, -a.x); }  // -i*a
__device__ __forceinline__ void c_fma(cplx& d, cplx a, cplx b) {            // d += a*b
  d.x = fmaf(a.x, b.x, fmaf(-a.y, b.y, d.x));
  d.y = fmaf(a.x, b.y, fmaf( a.y, b.x, d.y));
}
__device__ __forceinline__ void c_fma_conj(cplx& d, cplx a, cplx b) {       // d += conj(a)*b
  d.x = fmaf(a.x, b.x, fmaf( a.y, b.y, d.x));
  d.y = fmaf(a.x, b.y, fmaf(-a.y, b.x, d.y));
}

__device__ __forceinline__ void load_psi(const float4* __restrict__ p4, int s, cplx p[12]) {
  const float4* b = p4 + (size_t)s * 6;      // 24 floats = 6 x b128
#pragma unroll
  for (int i = 0; i < 6; ++i) {
    float4 v = b[i];
    p[2 * i]     = make_float2(v.x, v.y);
    p[2 * i + 1] = make_float2(v.z, v.w);
  }
}

// g = U h (two rhs at once): g[i] = sum_j U[i][j] h[j]
__device__ __forceinline__ void su3_mul(const cplx U[9], const cplx h0[3], const cplx h1[3],
                                        cplx g0[3], cplx g1[3]) {
#pragma unroll
  for (int i = 0; i < 3; ++i) {
    cplx a = make_float2(0.f, 0.f), b = make_float2(0.f, 0.f);
#pragma unroll
    for (int j = 0; j < 3; ++j) { c_fma(a, U[i * 3 + j], h0[j]); c_fma(b, U[i * 3 + j], h1[j]); }
    g0[i] = a; g1[i] = b;
  }
}
// g = U^dagger h: g[i] = sum_j conj(U[j][i]) h[j]
__device__ __forceinline__ void su3_dag_mul(const cplx U[9], const cplx h0[3], const cplx h1[3],
                                            cplx g0[3], cplx g1[3]) {
#pragma unroll
  for (int i = 0; i < 3; ++i) {
    cplx a = make_float2(0.f, 0.f), b = make_float2(0.f, 0.f);
#pragma unroll
    for (int j = 0; j < 3; ++j) { c_fma_conj(a, U[j * 3 + i], h0[j]); c_fma_conj(b, U[j * 3 + i], h1[j]); }
    g0[i] = a; g1[i] = b;
  }
}

#if defined(__HIP_DEVICE_COMPILE__) && defined(__gfx1250__) && \
    __has_builtin(__builtin_amdgcn_tensor_load_to_lds) &&      \
    __has_builtin(__builtin_amdgcn_s_wait_tensorcnt)
#define TDM_ENABLED 1
typedef uint32_t u32x4 __attribute__((ext_vector_type(4)));
typedef int32_t  i32x4 __attribute__((ext_vector_type(4)));
typedef int32_t  i32x8 __attribute__((ext_vector_type(8)));

// Issue one TDM load: tile_d1 rows of tile_d0 8-byte elements, row stride
// `stride0` elements, from global byte address `ga` to LDS byte offset `la`.
// Rows land contiguously in LDS (no padding).
__device__ __forceinline__ void tdm_load_2d(uint64_t ga, uint32_t la,
                                            int tile_d0, int tile_d1, int stride0) {
  // D# group0: count=1 | lds_addr | global_addr[31:0] | addr[56:32] + type=2
  u32x4 dg0 = { 1u, la, (uint32_t)ga,
                ((uint32_t)(ga >> 32) & 0x1FFFFFFu) | (2u << 30) };
  // D# group1: data_size=8B (code 3); tensor_dim0 = stride0 (full row, no OOB
  // clip); tensor_dim1 = tile_d1; tile_dim0/1; tensor_dim0_stride = stride0.
  i32x8 dg1 = { (int)(3u << 16),
                (stride0 & 0xFFFF) << 16,
                ((stride0 >> 16) & 0xFFFF) | ((tile_d1 & 0xFFFF) << 16),
                ((tile_d1 >> 16) & 0xFFFF) | ((tile_d0 & 0xFFFF) << 16),
                (tile_d1 & 0xFFFF),
                stride0, 0, 0 };
  i32x4 z4 = { 0, 0, 0, 0 };
  i32x8 z8 = { 0, 0, 0, 0, 0, 0, 0, 0 };
  // clang-23 / therock 6-arg form: (v4u, v8i, v4i, v4i, v8i, i32 cpol)
  __builtin_amdgcn_tensor_load_to_lds(dg0, dg1, z4, z4, z8, 0);
}
#else
#define TDM_ENABLED 0
#endif

__global__ __launch_bounds__(TPB) void wilson_dslash_eo(const float4* __restrict__ psi4,
                                                        const float2* __restrict__ Ug,
                                                        float4* __restrict__ out4) {
#if TDM_ENABLED
  __shared__ cplx shU [TPB * 36];  // main gauge tile: 128 sites x 4 mu (36 KB)
  __shared__ cplx shUt[TPB * 9];   // bwd-T links U_t(s - t^)   (9 KB)
  __shared__ cplx shUz[TPB * 9];   // bwd-Z links U_z(s - z^)   (9 KB)
  __shared__ cplx shUy[32  * 9];   // bwd-Y links for row-0 threads (2.25 KB)
#endif
  const int lsite = threadIdx.x;
  const int bs = blockIdx.x * TPB;
  const int s = bs + lsite;                        // bits[4:0]=x,[9:5]=y,[14:10]=z,[19:15]=t
  const int x = s & 31, y = (s >> 5) & 31, z = (s >> 10) & 31, t = (s >> 15) & 31;
  const int sxp = (s & ~31) | ((x + 1) & 31);
  const int sxm = (s & ~31) | ((x - 1) & 31);
  const int syp = (s & ~(31 << 5))  | (((y + 1) & 31) << 5);
  const int sym = (s & ~(31 << 5))  | (((y - 1) & 31) << 5);
  const int szp = (s & ~(31 << 10)) | (((z + 1) & 31) << 10);
  const int szm = (s & ~(31 << 10)) | (((z - 1) & 31) << 10);
  const int stp = (s & ~(31 << 15)) | (((t + 1) & 31) << 15);
  const int stm = (s & ~(31 << 15)) | (((t - 1) & 31) << 15);

  // Warm L2 with the neighbor spinors while the gauge DMAs are in flight.
  __builtin_prefetch(psi4 + (size_t)stp * 6, 0, 0);
  __builtin_prefetch(psi4 + (size_t)stm * 6, 0, 0);
  __builtin_prefetch(psi4 + (size_t)szp * 6, 0, 0);
  __builtin_prefetch(psi4 + (size_t)szm * 6, 0, 0);
  __builtin_prefetch(psi4 + (size_t)syp * 6, 0, 0);
  __builtin_prefetch(psi4 + (size_t)sym * 6, 0, 0);
  __builtin_prefetch(psi4 + (size_t)sxp * 6, 0, 0);
  __builtin_prefetch(psi4 + (size_t)sxm * 6, 0, 0);

#if TDM_ENABLED
  if (threadIdx.x < 32) {  // wave 0 issues the 4 gauge DMAs for the block
    // Blocks have uniform t,z and y_base (bs multiple of 128): the backward
    // T/Z/Y link sets are uniform contiguous-row strided tiles (wrap-safe).
    const int tB = (bs >> 15) & 31, zB = (bs >> 10) & 31, yB = (bs >> 5) & 31;
    const int tb = (tB > 0) ? bs - 32 * 1024 : bs + 31 * 32 * 1024;
    const int zb = (zB > 0) ? bs - 1024      : bs + 31 * 1024;
    const int yb = (yB > 0) ? bs - 32        : bs + 31 * 32;
    // main tile: one 4608-element contiguous row
    tdm_load_2d((uint64_t)(uintptr_t)(Ug + (size_t)bs * 36),
                (uint32_t)(uintptr_t)(&shU[0]), 4608, 1, 4608);
    // bwd-T: mu=0 link of 128 consecutive sites, stride 36
    tdm_load_2d((uint64_t)(uintptr_t)(Ug + (size_t)tb * 36 + 0 * 9),
                (uint32_t)(uintptr_t)(&shUt[0]), 9, 128, 36);
    // bwd-Z: mu=1 link of 128 consecutive sites
    tdm_load_2d((uint64_t)(uintptr_t)(Ug + (size_t)zb * 36 + 1 * 9),
                (uint32_t)(uintptr_t)(&shUz[0]), 9, 128, 36);
    // bwd-Y (row-0 threads only): mu=2 link of 32 consecutive sites
    tdm_load_2d((uint64_t)(uintptr_t)(Ug + (size_t)yb * 36 + 2 * 9),
                (uint32_t)(uintptr_t)(&shUy[0]), 9, 32, 36);
    __builtin_amdgcn_s_wait_tensorcnt(0);
  }
  __syncthreads();
#endif

  cplx Um[9];
#if TDM_ENABLED
#define LOAD_U_FWD(mu) do { const int o = (lsite * 4 + (mu)) * 9; \
    _Pragma("unroll") for (int k = 0; k < 9; ++k) Um[k] = shU[o + k]; } while (0)
#define LOAD_U_BWD_T() do { const int o = lsite * 9; \
    _Pragma("unroll") for (int k = 0; k < 9; ++k) Um[k] = shUt[o + k]; } while (0)
#define LOAD_U_BWD_Z() do { const int o = lsite * 9; \
    _Pragma("unroll") for (int k = 0; k < 9; ++k) Um[k] = shUz[o + k]; } while (0)
#define LOAD_U_BWD_Y() do { if (lsite >= 32) { const int o = ((lsite - 32) * 4 + 2) * 9; \
    _Pragma("unroll") for (int k = 0; k < 9; ++k) Um[k] = shU[o + k]; } \
    else { const int o = lsite * 9; \
    _Pragma("unroll") for (int k = 0; k < 9; ++k) Um[k] = shUy[o + k]; } } while (0)
#define LOAD_U_BWD_X() do { const int o = ((((lsite & ~31) | ((x - 1) & 31))) * 4 + 3) * 9; \
    _Pragma("unroll") for (int k = 0; k < 9; ++k) Um[k] = shU[o + k]; } while (0)
#else
#define LOAD_U_FWD(mu) do { const float2* ub = Ug + ((size_t)s * 4 + (mu)) * 9; \
    _Pragma("unroll") for (int k = 0; k < 9; ++k) Um[k] = ub[k]; } while (0)
#define LOAD_U_BWD_T() do { const float2* ub = Ug + ((size_t)stm * 4 + 0) * 9; \
    _Pragma("unroll") for (int k = 0; k < 9; ++k) Um[k] = ub[k]; } while (0)
#define LOAD_U_BWD_Z() do { const float2* ub = Ug + ((size_t)szm * 4 + 1) * 9; \
    _Pragma("unroll") for (int k = 0; k < 9; ++k) Um[k] = ub[k]; } while (0)
#define LOAD_U_BWD_Y() do { const float2* ub = Ug + ((size_t)sym * 4 + 2) * 9; \
    _Pragma("unroll") for (int k = 0; k < 9; ++k) Um[k] = ub[k]; } while (0)
#define LOAD_U_BWD_X() do { const float2* ub = Ug + ((size_t)sxm * 4 + 3) * 9; \
    _Pragma("unroll") for (int k = 0; k < 9; ++k) Um[k] = ub[k]; } while (0)
#endif

  cplx r[12];
#pragma unroll
  for (int k = 0; k < 12; ++k) r[k] = make_float2(0.f, 0.f);
  cplx p[12], h0[3], h1[3], g0[3], g1[3];

  // ---------- mu = 0 (T), gamma_t ----------
  load_psi(psi4, stp, p);                                   // (1-gt): h0=p0-p2, h1=p1-p3 -> (g0,g1,-g0,-g1)
#pragma unroll
  for (int c = 0; c < 3; ++c) { h0[c] = c_sub(p[c], p[6 + c]); h1[c] = c_sub(p[3 + c], p[9 + c]); }
  LOAD_U_FWD(0);
  su3_mul(Um, h0, h1, g0, g1);
#pragma unroll
  for (int c = 0; c < 3; ++c) { r[c] = c_add(r[c], g0[c]); r[3 + c] = c_add(r[3 + c], g1[c]);
                                r[6 + c] = c_sub(r[6 + c], g0[c]); r[9 + c] = c_sub(r[9 + c], g1[c]); }
  load_psi(psi4, stm, p);                                   // (1+gt): h0=p0+p2, h1=p1+p3 -> (g0,g1,g0,g1)
#pragma unroll
  for (int c = 0; c < 3; ++c) { h0[c] = c_add(p[c], p[6 + c]); h1[c] = c_add(p[3 + c], p[9 + c]); }
  LOAD_U_BWD_T();
  su3_dag_mul(Um, h0, h1, g0, g1);
#pragma unroll
  for (int c = 0; c < 3; ++c) { r[c] = c_add(r[c], g0[c]); r[3 + c] = c_add(r[3 + c], g1[c]);
                                r[6 + c] = c_add(r[6 + c], g0[c]); r[9 + c] = c_add(r[9 + c], g1[c]); }

  // ---------- mu = 1 (Z), gamma_z ----------
  load_psi(psi4, szp, p);                                   // (1-gz): h0=p0-i*p2, h1=p1+i*p3 -> (g0,g1,i*g0,-i*g1)
#pragma unroll
  for (int c = 0; c < 3; ++c) { h0[c] = c_sub(p[c], c_i(p[6 + c])); h1[c] = c_add(p[3 + c], c_i(p[9 + c])); }
  LOAD_U_FWD(1);
  su3_mul(Um, h0, h1, g0, g1);
#pragma unroll
  for (int c = 0; c < 3; ++c) { r[c] = c_add(r[c], g0[c]); r[3 + c] = c_add(r[3 + c], g1[c]);
                                r[6 + c] = c_add(r[6 + c], c_i(g0[c])); r[9 + c] = c_add(r[9 + c], c_mi(g1[c])); }
  load_psi(psi4, szm, p);                                   // (1+gz): h0=p0+i*p2, h1=p1-i*p3 -> (g0,g1,-i*g0,i*g1)
#pragma unroll
  for (int c = 0; c < 3; ++c) { h0[c] = c_add(p[c], c_i(p[6 + c])); h1[c] = c_sub(p[3 + c], c_i(p[9 + c])); }
  LOAD_U_BWD_Z();
  su3_dag_mul(Um, h0, h1, g0, g1);
#pragma unroll
  for (int c = 0; c < 3; ++c) { r[c] = c_add(r[c], g0[c]); r[3 + c] = c_add(r[3 + c], g1[c]);
                                r[6 + c] = c_add(r[6 + c], c_mi(g0[c])); r[9 + c] = c_add(r[9 + c], c_i(g1[c])); }

  // ---------- mu = 2 (Y), gamma_y ----------
  load_psi(psi4, syp, p);                                   // (1-gy): h0=p0+p3, h1=p1-p2 -> (g0,g1,-g1,g0)
#pragma unroll
  for (int c = 0; c < 3; ++c) { h0[c] = c_add(p[c], p[9 + c]); h1[c] = c_sub(p[3 + c], p[6 + c]); }
  LOAD_U_FWD(2);
  su3_mul(Um, h0, h1, g0, g1);
#pragma unroll
  for (int c = 0; c < 3; ++c) { r[c] = c_add(r[c], g0[c]); r[3 + c] = c_add(r[3 + c], g1[c]);
                                r[6 + c] = c_sub(r[6 + c], g1[c]); r[9 + c] = c_add(r[9 + c], g0[c]); }
  load_psi(psi4, sym, p);                                   // (1+gy): h0=p0-p3, h1=p1+p2 -> (g0,g1,g1,-g0)
#pragma unroll
  for (int c = 0; c < 3; ++c) { h0[c] = c_sub(p[c], p[9 + c]); h1[c] = c_add(p[3 + c], p[6 + c]); }
  LOAD_U_BWD_Y();
  su3_dag_mul(Um, h0, h1, g0, g1);
#pragma unroll
  for (int c = 0; c < 3; ++c) { r[c] = c_add(r[c], g0[c]); r[3 + c] = c_add(r[3 + c], g1[c]);
                                r[6 + c] = c_add(r[6 + c], g1[c]); r[9 + c] = c_sub(r[9 + c], g0[c]); }

  // ---------- mu = 3 (X), gamma_x ----------
  load_psi(psi4, sxp, p);                                   // (1-gx): h0=p0-i*p3, h1=p1-i*p2 -> (g0,g1,i*g1,i*g0)
#pragma unroll
  for (int c = 0; c < 3; ++c) { h0[c] = c_sub(p[c], c_i(p[9 + c])); h1[c] = c_sub(p[3 + c], c_i(p[6 + c])); }
  LOAD_U_FWD(3);
  su3_mul(Um, h0, h1, g0, g1);
#pragma unroll
  for (int c = 0; c < 3; ++c) { r[c] = c_add(r[c], g0[c]); r[3 + c] = c_add(r[3 + c], g1[c]);
                                r[6 + c] = c_add(r[6 + c], c_i(g1[c])); r[9 + c] = c_add(r[9 + c], c_i(g0[c])); }
  load_psi(psi4, sxm, p);                                   // (1+gx): h0=p0+i*p3, h1=p1+i*p2 -> (g0,g1,-i*g1,-i*g0)
#pragma unroll
  for (int c = 0; c < 3; ++c) { h0[c] = c_add(p[c], c_i(p[9 + c])); h1[c] = c_add(p[3 + c], c_i(p[6 + c])); }
  LOAD_U_BWD_X();
  su3_dag_mul(Um, h0, h1, g0, g1);
#pragma unroll
  for (int c = 0; c < 3; ++c) { r[c] = c_add(r[c], g0[c]); r[3 + c] = c_add(r[3 + c], g1[c]);
                                r[6 + c] = c_add(r[6 + c], c_mi(g1[c])); r[9 + c] = c_add(r[9 + c], c_mi(g0[c])); }

  // out = -0.5 * r, packed as 6 x b128 stores
  float4* ob = out4 + (size_t)s * 6;
#pragma unroll
  for (int i = 0; i < 6; ++i) {
    cplx a = r[2 * i], b = r[2 * i + 1];
    ob[i] = make_float4(-0.5f * a.x, -0.5f * a.y, -0.5f * b.x, -0.5f * b.y);
  }
}

extern "C" void kernel_launch(void* const* d_in, const int* in_sizes, int n_in,
                              void* d_out, int out_size, void* d_ws, size_t ws_size,
                              hipStream_t stream) {
  (void)in_sizes; (void)n_in; (void)out_size; (void)d_ws; (void)ws_size;
  const float4* psi = (const float4*)d_in[0];  // complex64 spinor field, 6 float4 per site
  const float2* U   = (const float2*)d_in[1];  // complex64 gauge field, 36 float2 per site
  float4* out = (float4*)d_out;
  wilson_dslash_eo<<<VOL / TPB, TPB, 0, stream>>>(psi, U, out);
}